// YOLOLoss_48550310314251
// MI455X (gfx1250) — compile-verified
//
#include <hip/hip_runtime.h>

// ---------------------------------------------------------------------------
// YOLOv3 loss for gfx1250 (MI455X).
//  - 3 layers, each: one thread per (anchor, j, i) cell; targets precomputed
//    into LDS per block (threads 0..T-1).
//  - Only channels 0..4 are streamed for every cell; class channels (5..84)
//    are read only for the <=50 assigned cells per (batch, layer).
//  - Wave reduction via V_WMMA_F32_16X16X4_F32 (B = ones), block reduction
//    via LDS, global accumulation via global_atomic_add_f32.
// ---------------------------------------------------------------------------

#define A_PER 3
#define NCH   85
#define NCLS  80
#define T_MAX 50

__constant__ float c_anchors[18] = {
    10.f, 13.f,  16.f, 30.f,  33.f, 23.f,
    30.f, 61.f,  62.f, 45.f,  59.f, 119.f,
    116.f, 90.f, 156.f, 198.f, 373.f, 326.f};

typedef __attribute__((ext_vector_type(2))) float v2f;
typedef __attribute__((ext_vector_type(8))) float v8f;

__device__ __forceinline__ float safe_logf(float x) {
  return (x > 0.0f) ? logf(x) : -100.0f;
}

__device__ __forceinline__ float bce(float o, float t) {
  return -(t * safe_logf(o) + (1.0f - t) * safe_logf(1.0f - o));
}

// Sum of a value across all 32 lanes of a wave, returned to every lane.
// Uses the CDNA5 matrix pipe: D = A x ones. A is 16x4 f32 (2 VGPRs/lane):
// lane L<16 holds A[L,0..1], lane L>=16 holds A[L-16,2..3]. With the partial
// in component 0 and 0 in component 1, row m of D equals x[m] + x[m+16] in
// every column. Each lane then holds 8 rows of its column in d[0..7];
// summing them gives rows 0..7 (lanes 0..15) or rows 8..15 (lanes 16..31),
// and one xor-16 shuffle completes the total.
__device__ __forceinline__ float wave_sum(float x) {
#if defined(__AMDGCN__) && __has_builtin(__builtin_amdgcn_wmma_f32_16x16x4_f32)
  v2f a; a[0] = x;    a[1] = 0.0f;
  v2f b; b[0] = 1.0f; b[1] = 1.0f;
  v8f c = {};
  v8f d = __builtin_amdgcn_wmma_f32_16x16x4_f32(
      /*neg_a=*/false, a, /*neg_b=*/false, b,
      /*c_mod=*/(short)0, c, /*reuse_a=*/false, /*reuse_b=*/false);
  float s = d[0] + d[1] + d[2] + d[3] + d[4] + d[5] + d[6] + d[7];
  s += __shfl_xor(s, 16, 32);
  return s;
#else
  for (int off = 16; off > 0; off >>= 1) x += __shfl_xor(x, off, 32);
  return x;
#endif
}

__global__ void zero_scalar(float* out) {
  if (threadIdx.x == 0 && blockIdx.x == 0) out[0] = 0.0f;
}

template <int HH, int STRIDE, int GROUP>
__global__ __launch_bounds__(256) void yolo_layer_loss(
    const float* __restrict__ raw,      // (B, 255, HH, HH)
    const float* __restrict__ targets,  // (B, T, 5): cls, x, y, w, h
    float* __restrict__ out, int T) {
  constexpr int WW = HH;
  constexpr int NCELL = A_PER * HH * WW;
  constexpr float inv_stride = 1.0f / (float)STRIDE;
  const int b = blockIdx.y;
  const int tid = threadIdx.x;
  const float fsize = (float)HH;

  // Per-target record:
  // 0:valid 1:tx 2:ty 3:tw 4:th 5:a_sel(-1 if none) 6:gi 7:gj
  // 8:dx 9:dy 10:log(tw/aw) 11:log(th/ah) 12:scale 13:cls
  __shared__ float tg[T_MAX][16];
  __shared__ float wave_part[8];

  if (tid < T) {
    const float* lab = targets + ((size_t)b * T + tid) * 5;
    float cls = lab[0], lx = lab[1], ly = lab[2], lw = lab[3], lh = lab[4];
    float valid = (cls + lx + ly + lw + lh) > 0.0f ? 1.0f : 0.0f;
    float tx = lx * fsize, ty = ly * fsize, tw = lw * fsize, th = lh * fsize;
    float barea = tw * th;
    float bv = -1.0f;
    int bk = 0;
#pragma unroll
    for (int k = 0; k < 9; ++k) {
      float agw = c_anchors[2 * k] * inv_stride;
      float agh = c_anchors[2 * k + 1] * inv_stride;
      float mw = fminf(tw, agw), mh = fminf(th, agh);
      float inter = (mw > 0.0f && mh > 0.0f) ? mw * mh : 0.0f;
      float iou = inter / (barea + agw * agh - inter);
      if (iou > bv) { bv = iou; bk = k; }  // first max, like jnp.argmax
    }
    int best_n = bk - (bk / 3) * 3;
    bool sel = (valid > 0.0f) && ((bk / 3) == GROUP);
    int gi = (int)tx, gj = (int)ty;
    float aw = c_anchors[2 * (3 * GROUP + best_n)] * inv_stride;
    float ah = c_anchors[2 * (3 * GROUP + best_n) + 1] * inv_stride;
    tg[tid][0] = valid;
    tg[tid][1] = tx;  tg[tid][2] = ty;  tg[tid][3] = tw;  tg[tid][4] = th;
    tg[tid][5] = sel ? (float)best_n : -1.0f;
    tg[tid][6] = (float)gi;
    tg[tid][7] = (float)gj;
    tg[tid][8] = tx - (float)gi;
    tg[tid][9] = ty - (float)gj;
    tg[tid][10] = logf(tw / aw + 1e-16f);
    tg[tid][11] = logf(th / ah + 1e-16f);
    tg[tid][12] = sqrtf(2.0f - tw * th / (fsize * fsize));
    tg[tid][13] = cls;
  }
  __syncthreads();

  const int cell = blockIdx.x * blockDim.x + tid;
  float loss = 0.0f;
  if (cell < NCELL) {
    const int a = cell / (HH * WW);
    const int rem = cell - a * (HH * WW);
    const int j = rem / WW;
    const int i = rem - j * WW;
    const size_t cs = (size_t)HH * WW;  // channel stride
    const float* base =
        raw + ((size_t)b * (A_PER * NCH) + (size_t)a * NCH) * cs + (size_t)j * WW + i;

    float o0 = base[0];
    float o1 = base[cs];
    float o2 = base[2 * cs];
    float o3 = base[3 * cs];
    float o4 = base[4 * cs];

    float sx = 1.0f / (1.0f + expf(-o0));
    float sy = 1.0f / (1.0f + expf(-o1));
    float sob = 1.0f / (1.0f + expf(-o4));

    const float maw = c_anchors[2 * (3 * GROUP + a)] * inv_stride;
    const float mah = c_anchors[2 * (3 * GROUP + a) + 1] * inv_stride;
    float px = sx + (float)i;
    float py = sy + (float)j;
    float pw = expf(o2) * maw;
    float ph = expf(o3) * mah;
    float ax0 = px - pw * 0.5f, ay0 = py - ph * 0.5f;
    float ax1 = px + pw * 0.5f, ay1 = py + ph * 0.5f;
    float area_a = pw * ph;

    float max_iou = 0.0f;
    int asg = -1;  // last matching target wins (scatter duplicate order)
    for (int t = 0; t < T; ++t) {
      if (tg[t][0] > 0.0f) {
        float tx = tg[t][1], ty = tg[t][2], tw = tg[t][3], th = tg[t][4];
        float hw = tw * 0.5f, hh = th * 0.5f;
        float tlx = fmaxf(ax0, tx - hw), tly = fmaxf(ay0, ty - hh);
        float brx = fminf(ax1, tx + hw), bry = fminf(ay1, ty + hh);
        if (tlx < brx && tly < bry) {
          float ai = (brx - tlx) * (bry - tly);
          float iou = ai / (area_a + tw * th - ai);
          max_iou = fmaxf(max_iou, iou);
        }
        if (tg[t][5] == (float)a && tg[t][6] == (float)i && tg[t][7] == (float)j)
          asg = t;
      }
    }

    if (asg >= 0) {
      // Assigned cell: xy/wh/obj/cls losses; obj_mask forced to 1.
      float ts = tg[asg][12];
      float w2 = ts * ts;
      loss += w2 * (bce(sx, tg[asg][8]) + bce(sy, tg[asg][9]));
      float dw = (o2 - tg[asg][10]) * ts;
      float dh = (o3 - tg[asg][11]) * ts;
      loss += 0.5f * (dw * dw + dh * dh);
      loss += bce(sob, 1.0f);
      const int cls = (int)tg[asg][13];
      for (int c = 0; c < NCLS; ++c) {
        float oc = base[(size_t)(5 + c) * cs];
        float sc = 1.0f / (1.0f + expf(-oc));
        loss += bce(sc, (c == cls) ? 1.0f : 0.0f);
      }
    } else if (!(max_iou > 0.7f)) {
      // Unassigned, not ignored: objectness-vs-0 only.
      loss += bce(sob, 0.0f);
    }
  }

  // Wave reduction on the CDNA5 matrix pipe, then block reduce via LDS.
  float wsum = wave_sum(loss);
  const int lane = tid & 31, wv = tid >> 5;
  if (lane == 0) wave_part[wv] = wsum;
  __syncthreads();
  if (tid == 0) {
    float s = 0.0f;
    const int nw = (int)((blockDim.x + 31) >> 5);
    for (int k = 0; k < nw; ++k) s += wave_part[k];
    atomicAdd(out, s);
  }
}

extern "C" void kernel_launch(void* const* d_in, const int* in_sizes, int n_in,
                              void* d_out, int out_size, void* d_ws, size_t ws_size,
                              hipStream_t stream) {
  (void)n_in; (void)out_size; (void)d_ws; (void)ws_size;
  const float* out0 = (const float*)d_in[0];  // (B,255,19,19)
  const float* out1 = (const float*)d_in[1];  // (B,255,38,38)
  const float* out2 = (const float*)d_in[2];  // (B,255,76,76)
  const float* tgts = (const float*)d_in[3];  // (B,50,5)
  float* out = (float*)d_out;

  const int T = T_MAX;
  int B = in_sizes[3] / (T * 5);
  if (B < 1) B = 1;

  zero_scalar<<<1, 32, 0, stream>>>(out);

  constexpr int BLK = 256;
  {
    const int ncell = A_PER * 19 * 19;
    dim3 g((ncell + BLK - 1) / BLK, B);
    yolo_layer_loss<19, 32, 2><<<g, BLK, 0, stream>>>(out0, tgts, out, T);
  }
  {
    const int ncell = A_PER * 38 * 38;
    dim3 g((ncell + BLK - 1) / BLK, B);
    yolo_layer_loss<38, 16, 1><<<g, BLK, 0, stream>>>(out1, tgts, out, T);
  }
  {
    const int ncell = A_PER * 76 * 76;
    dim3 g((ncell + BLK - 1) / BLK, B);
    yolo_layer_loss<76, 8, 0><<<g, BLK, 0, stream>>>(out2, tgts, out, T);
  }
}